// Model_34780645163600
// MI455X (gfx1250) — compile-verified
//
#include <hip/hip_runtime.h>

// ---------------------------------------------------------------------------
// MI455X / gfx1250 implementation.
// Bandwidth-bound pipeline (~42 GFLOP fp32 vs ~900MB intermediate traffic at
// 23.3 TB/s), so we keep full f32 precision and use V_WMMA_F32_16X16X4_F32
// (wave32, M=16,N=16,K=4) for conv2/conv3/FC implicit GEMMs.
// conv3 (dominant: ~28 GFLOP) stages its 3 input rows (24KB, contiguous) into
// LDS with the Tensor Data Mover (tensor_load_to_lds + s_wait_tensorcnt),
// shared by all 8 waves of the block; A-fragments then come from LDS.
//
// Workspace layout (floats), peak ~694 MB:
//   f3   @ 0           : 256*94*126*32 = 97,026,048 (+2,048 pad)
//   f1   @ 97,028,096  : 256*98*130*8  = 26,091,520
//   f2   @ 123,119,616 : 256*96*128*16 = 50,331,648
//   A    @ f2 offset   (f2 dead after conv3)  : 256*94*60*32 = 46,202,880
//   p    @ f1 offset   (f1 dead after conv2)  : 125*44*60*32 = 10,560,000
//   part @ f1+10,560,000 : 32*256*32 = 262,144
//   h    @ part+262,144  : 256*32    = 8,192
//   ys   @ h+8,192       : 32*256*32 = 262,144
// ---------------------------------------------------------------------------

typedef __attribute__((ext_vector_type(2))) float        v2f;
typedef __attribute__((ext_vector_type(8))) float        v8f;
typedef __attribute__((ext_vector_type(4))) unsigned int u32x4;
typedef __attribute__((ext_vector_type(4))) int          i32x4;
typedef __attribute__((ext_vector_type(8))) int          i32x8;

__device__ __forceinline__ v8f wmma_f32(v2f a, v2f b, v8f c) {
    // D = A(16x4) * B(4x16) + C(16x16), full f32. 8-arg form:
    // (neg_a, A, neg_b, B, c_mod, C, reuse_a, reuse_b)
    return __builtin_amdgcn_wmma_f32_16x16x4_f32(false, a, false, b,
                                                 (short)0, c, false, false);
}

// TDM: DMA a 2D f32 tile (rows x cols, row stride = cols) from global to LDS.
// D# group0/group1 packed per CDNA5 ISA sec 8.3/8.4. Issued by one wave,
// tracked with TENSORcnt. Caller must barrier before other waves read LDS.
__device__ __forceinline__ void tdm_load_2d_f32(unsigned lds_byte_addr,
                                                const float* src,
                                                unsigned cols, unsigned rows) {
    unsigned long long ga = (unsigned long long)(size_t)src;
    u32x4 g0;
    g0[0] = 1u;                                        // count=1, user mode
    g0[1] = lds_byte_addr;                             // lds_addr [63:32]
    g0[2] = (unsigned)(ga & 0xFFFFFFFFu);              // global_addr [95:64]
    g0[3] = (unsigned)((ga >> 32) & 0x01FFFFFFu)       // global_addr [120:96]
          | (2u << 30);                                // type=2 ("image")
    i32x8 g1;
    g1[0] = (int)(2u << 16);                           // data_size=2 (4 bytes)
    g1[1] = (int)((cols & 0xFFFFu) << 16);             // tensor_dim0 lo16
    g1[2] = (int)((cols >> 16) | ((rows & 0xFFFFu) << 16)); // td0 hi | td1 lo
    g1[3] = (int)((rows >> 16) | ((cols & 0xFFFFu) << 16)); // td1 hi | tile_dim0
    g1[4] = (int)(rows & 0xFFFFu);                     // tile_dim1, tile_dim2=0
    g1[5] = (int)cols;                                 // tensor_dim0_stride lo32
    g1[6] = (int)((cols & 0xFFFFu) << 16);             // td0s hi=0 | td1_stride lo16
    g1[7] = (int)(cols >> 16);                         // td1_stride hi
    i32x4 z4 = {0, 0, 0, 0};
#if defined(__clang_major__) && (__clang_major__ >= 23)
    i32x8 z8 = {0, 0, 0, 0, 0, 0, 0, 0};
    __builtin_amdgcn_tensor_load_to_lds(g0, g1, z4, z4, z8, 0);
#else
    __builtin_amdgcn_tensor_load_to_lds(g0, g1, z4, z4, 0);
#endif
    __builtin_amdgcn_s_wait_tensorcnt(0);
}

// ---------------- conv1: (256,100,132,2) -> (256,98,130,8), K=18 (VALU) ----
__global__ void conv1_kernel(const float* __restrict__ x,
                             const float* __restrict__ w1,
                             const float* __restrict__ b1,
                             float* __restrict__ f1) {
    int idx = blockIdx.x * 256 + threadIdx.x;              // 26,091,520 total
    if (idx >= 256 * 98 * 130 * 8) return;
    int co = idx & 7;
    int xw = (idx >> 3) % 130;
    int y  = ((idx >> 3) / 130) % 98;
    int t  = idx / (8 * 130 * 98);
    float acc = b1[co];
    #pragma unroll
    for (int dy = 0; dy < 3; ++dy)
        #pragma unroll
        for (int dx = 0; dx < 3; ++dx) {
            const float* px = x + (((t * 100) + y + dy) * 132 + (xw + dx)) * 2;
            const float* pw = w1 + ((dy * 3 + dx) * 2) * 8 + co;
            acc += px[0] * pw[0] + px[1] * pw[8];
        }
    f1[idx] = acc;
}

// ---------------- conv2: (256,98,130,8) -> (256,96,128,16), WMMA ------------
// One wave per tile: 16 output pixels (along W) x 16 out channels.
// 9 taps x 2 channel-groups-of-4 = 18 WMMAs per tile.
__global__ void conv2_kernel(const float* __restrict__ f1,
                             const float* __restrict__ w2,
                             const float* __restrict__ b2,
                             float* __restrict__ f2) {
    int wid  = blockIdx.x * 8 + (threadIdx.x >> 5);        // 196,608 waves
    int lane = threadIdx.x & 31;
    int xt = wid & 7;
    int y  = (wid >> 3) % 96;
    int t  = wid / (8 * 96);
    int x0 = xt * 16;
    int m  = lane & 15;        // A row (pixel) / B,D column (out channel)
    int kh = lane >> 4;        // K half-select per ISA A/B layout

    v8f acc = {0.f, 0.f, 0.f, 0.f, 0.f, 0.f, 0.f, 0.f};
    for (int dy = 0; dy < 3; ++dy)
        for (int dx = 0; dx < 3; ++dx) {
            int rowbase = ((t * 98 + y + dy) * 130 + (x0 + m + dx)) * 8;
            #pragma unroll
            for (int cg = 0; cg < 2; ++cg) {
                int ai = rowbase + cg * 4 + 2 * kh;
                v2f a; a.x = f1[ai]; a.y = f1[ai + 1];
                int bi = ((dy * 3 + dx) * 8 + cg * 4 + 2 * kh) * 16 + m;
                v2f b; b.x = w2[bi]; b.y = w2[bi + 16];
                acc = wmma_f32(a, b, acc);
            }
        }
    float bias = b2[m];
    #pragma unroll
    for (int j = 0; j < 8; ++j) {
        int xr = x0 + j + 8 * kh;                           // D row
        f2[((t * 96 + y) * 128 + xr) * 16 + m] = acc[j] + bias;
    }
}

// ---------------- conv3: (256,96,128,16) -> (256,94,126,32), WMMA + TDM -----
// Block = one (t,y) strip: 8 waves x (16 pixels x 32 out channels).
// Input rows y..y+2 (3 x 128 x 16 f32 = 24KB, contiguous) are DMA'd into LDS
// by the Tensor Data Mover once per block, then all taps read from LDS.
// 9 taps x 4 cgroups x 2 N-tiles = 72 WMMAs per wave.
__global__ void conv3_kernel(const float* __restrict__ f2,
                             const float* __restrict__ w3,
                             const float* __restrict__ b3,
                             float* __restrict__ f3) {
    __shared__ float smem[3 * 2048];                       // 24KB per block
    int blk  = blockIdx.x;                                 // 24,064 blocks
    int y    = blk % 94;
    int t    = blk / 94;
    int lane = threadIdx.x & 31;
    int xt   = threadIdx.x >> 5;                           // 0..7 (x tile)
    int x0   = xt * 16;
    int m    = lane & 15;
    int kh   = lane >> 4;

    if (threadIdx.x < 32) {
        // One wave issues the TDM load of rows y..y+2 (EXEC ignored by TDM).
        tdm_load_2d_f32((unsigned)(size_t)(void*)smem,
                        f2 + (size_t)(t * 96 + y) * 2048, 2048u, 3u);
    }
    __syncthreads();

    v8f acc0 = {0.f, 0.f, 0.f, 0.f, 0.f, 0.f, 0.f, 0.f};
    v8f acc1 = {0.f, 0.f, 0.f, 0.f, 0.f, 0.f, 0.f, 0.f};
    for (int dy = 0; dy < 3; ++dy)
        for (int dx = 0; dx < 3; ++dx) {
            int xi = x0 + m + dx;
            if (xi > 127) xi = 127;                        // clamp edge loads
            int rowbase = dy * 2048 + xi * 16;
            #pragma unroll
            for (int cg = 0; cg < 4; ++cg) {
                int ai = rowbase + cg * 4 + 2 * kh;
                v2f a; a.x = smem[ai]; a.y = smem[ai + 1]; // ds_load_b64
                int bi = ((dy * 3 + dx) * 16 + cg * 4 + 2 * kh) * 32 + m;
                v2f b0; b0.x = w3[bi];      b0.y = w3[bi + 32];
                v2f b1; b1.x = w3[bi + 16]; b1.y = w3[bi + 48];
                acc0 = wmma_f32(a, b0, acc0);
                acc1 = wmma_f32(a, b1, acc1);
            }
        }
    float bias0 = b3[m], bias1 = b3[m + 16];
    #pragma unroll
    for (int j = 0; j < 8; ++j) {
        int xr = x0 + j + 8 * kh;
        if (xr < 126) {                                    // predicate stores
            int o = ((t * 94 + y) * 126 + xr) * 32;
            f3[o + m]      = acc0[j] + bias0;
            f3[o + 16 + m] = acc1[j] + bias1;
        }
    }
}

// ---------------- separable avg pool -----------------------------------------
// Stage A: sum over W window 8 stride 2: (256,94,126,32) -> (256,94,60,32)
__global__ void poolA_kernel(const float* __restrict__ f3,
                             float* __restrict__ A) {
    int idx = blockIdx.x * 256 + threadIdx.x;              // 46,202,880 total
    if (idx >= 256 * 94 * 60 * 32) return;
    int c  = idx & 31;
    int xw = (idx >> 5) % 60;
    int y  = ((idx >> 5) / 60) % 94;
    int t  = idx / (32 * 60 * 94);
    int base = ((t * 94 + y) * 126 + 2 * xw) * 32 + c;
    float s = 0.f;
    #pragma unroll
    for (int i = 0; i < 8; ++i) s += f3[base + i * 32];
    A[idx] = s;
}

// Stage B: sum over T,H windows 8 stride 2, /512: -> (125,44,60,32)
__global__ void poolB_kernel(const float* __restrict__ A,
                             float* __restrict__ p) {
    int idx = blockIdx.x * 256 + threadIdx.x;              // 10,560,000 total
    if (idx >= 125 * 44 * 60 * 32) return;
    int c  = idx & 31;
    int xw = (idx >> 5) % 60;
    int yy = ((idx >> 5) / 60) % 44;
    int tt = idx / (32 * 60 * 44);
    float s = 0.f;
    for (int a = 0; a < 8; ++a)
        #pragma unroll
        for (int b = 0; b < 8; ++b)
            s += A[(((2 * tt + a) * 94 + (2 * yy + b)) * 60 + xw) * 32 + c];
    p[idx] = s * (1.f / 512.f);
}

// ---------------- FC: (256,41250)@(41250,32), WMMA split-K -------------------
// 16 M-tiles x 2 N-tiles x 32 K-splits = 1024 waves; K padded (zero fill) to a
// multiple of 4; deterministic two-pass reduction (no float atomics).
__global__ void fc_kernel(const float* __restrict__ p,
                          const float* __restrict__ w_in,
                          float* __restrict__ partials) {
    int wid  = blockIdx.x * 8 + (threadIdx.x >> 5);        // 0..1023
    int lane = threadIdx.x & 31;
    int ks = wid & 31;
    int nt = (wid >> 5) & 1;
    int mt = wid >> 6;
    int m  = lane & 15;
    int kh = lane >> 4;
    const int K = 41250, KK = 10313, CH = 323;             // ceil(K/4), ceil(KK/32)
    int k0 = ks * CH;
    int k1 = k0 + CH; if (k1 > KK) k1 = KK;
    int r = mt * 16 + m;
    int n = nt * 16 + m;
    const float* prow = p + (long)r * K;

    v8f acc = {0.f, 0.f, 0.f, 0.f, 0.f, 0.f, 0.f, 0.f};
    for (int kk = k0; kk < k1; ++kk) {
        int kb = 4 * kk + 2 * kh;                          // even; kb<K => kb+1<K
        v2f a, b;
        if (kb < K) {
            a.x = prow[kb];            a.y = prow[kb + 1];
            b.x = w_in[kb * 32 + n];   b.y = w_in[(kb + 1) * 32 + n];
        } else {
            a.x = 0.f; a.y = 0.f; b.x = 0.f; b.y = 0.f;
        }
        acc = wmma_f32(a, b, acc);
    }
    #pragma unroll
    for (int j = 0; j < 8; ++j) {
        int rr = mt * 16 + j + 8 * kh;
        partials[((long)ks * 256 + rr) * 32 + nt * 16 + m] = acc[j];
    }
}

__global__ void fc_reduce_kernel(const float* __restrict__ partials,
                                 const float* __restrict__ b_in,
                                 float* __restrict__ h) {
    int idx = blockIdx.x * 256 + threadIdx.x;              // 8192
    if (idx >= 8192) return;
    float s = b_in[idx & 31];
    for (int ss = 0; ss < 32; ++ss) s += partials[ss * 8192 + idx];
    h[idx] = s;
}

// ---------------- LIF RNN scan (single 1024-thread block, LDS resident) -----
// xs[t,i] = h_flat[i*256+t] (the reshape+transpose in the reference).
// Thread (b,n) owns V1..Vl[b,n]; 32x32 matmuls via LDS spike buffer.
// ys stored directly in sp layout: ys[b*8192 + t*32 + n].
__global__ __launch_bounds__(1024) void rnn_kernel(
    const float* __restrict__ h,
    const float* __restrict__ beta1, const float* __restrict__ wl1, const float* __restrict__ bl1,
    const float* __restrict__ beta2, const float* __restrict__ wl2, const float* __restrict__ bl2,
    const float* __restrict__ beta3, const float* __restrict__ wl3, const float* __restrict__ bl3,
    const float* __restrict__ beta_li,
    float* __restrict__ ys) {
    __shared__ float sw1[1024], sw2[1024], sw3[1024];
    __shared__ float sh[8192];
    __shared__ float sb[1024];
    int tid = threadIdx.x;
    int b = tid >> 5, n = tid & 31;
    sw1[tid] = wl1[tid]; sw2[tid] = wl2[tid]; sw3[tid] = wl3[tid];
    #pragma unroll
    for (int i = 0; i < 8; ++i) sh[tid + i * 1024] = h[tid + i * 1024];
    float bb1 = fminf(fmaxf(beta1[n],   0.f), 1.f);
    float bb2 = fminf(fmaxf(beta2[n],   0.f), 1.f);
    float bb3 = fminf(fmaxf(beta3[n],   0.f), 1.f);
    float bbl = fminf(fmaxf(beta_li[n], 0.f), 1.f);
    float c1 = bl1[n], c2 = bl2[n], c3 = bl3[n];
    float V1 = 0.f, V2 = 0.f, V3 = 0.f, Vl = 0.f;
    __syncthreads();
    for (int t = 0; t < 256; ++t) {
        float xt = sh[n * 256 + t];
        float s1 = (V1 - 1.f > 0.f) ? 1.f : 0.f;
        __syncthreads(); sb[tid] = s1; __syncthreads();
        float h1 = c1;
        #pragma unroll
        for (int j = 0; j < 32; ++j) h1 += sb[b * 32 + j] * sw1[j * 32 + n];
        V1 = bb1 * V1 + xt - s1;
        float s2 = (V2 - 1.f > 0.f) ? 1.f : 0.f;
        __syncthreads(); sb[tid] = s2; __syncthreads();
        float h2 = c2;
        #pragma unroll
        for (int j = 0; j < 32; ++j) h2 += sb[b * 32 + j] * sw2[j * 32 + n];
        V2 = bb2 * V2 + h1 - s2;
        float s3 = (V3 - 1.f > 0.f) ? 1.f : 0.f;
        __syncthreads(); sb[tid] = s3; __syncthreads();
        float h3 = c3;
        #pragma unroll
        for (int j = 0; j < 32; ++j) h3 += sb[b * 32 + j] * sw3[j * 32 + n];
        V3 = bb3 * V3 + h2 - s3;
        Vl = bbl * Vl + h3;
        ys[b * 8192 + t * 32 + n] = Vl;
    }
}

// ---------------- final projection: (256,1024)@(1024,2) ---------------------
__global__ void out_kernel(const float* __restrict__ ys,
                           const float* __restrict__ w_out,
                           const float* __restrict__ b_out,
                           float* __restrict__ out) {
    int idx = blockIdx.x * 256 + threadIdx.x;              // 512
    if (idx >= 512) return;
    int o = idx & 1, r = idx >> 1;
    float s = b_out[o];
    const float* row = ys + r * 1024;
    for (int c = 0; c < 1024; ++c) s += row[c] * w_out[c * 2 + o];
    out[idx] = s;
}

// ---------------------------------------------------------------------------
extern "C" void kernel_launch(void* const* d_in, const int* in_sizes, int n_in,
                              void* d_out, int out_size, void* d_ws, size_t ws_size,
                              hipStream_t stream) {
    const float* x       = (const float*)d_in[0];
    const float* w1      = (const float*)d_in[1];
    const float* b1      = (const float*)d_in[2];
    const float* w2      = (const float*)d_in[3];
    const float* b2      = (const float*)d_in[4];
    const float* w3      = (const float*)d_in[5];
    const float* b3      = (const float*)d_in[6];
    const float* w_in    = (const float*)d_in[7];
    const float* b_in    = (const float*)d_in[8];
    const float* beta1   = (const float*)d_in[9];
    const float* wl1     = (const float*)d_in[10];
    const float* bl1     = (const float*)d_in[11];
    const float* beta2   = (const float*)d_in[12];
    const float* wl2     = (const float*)d_in[13];
    const float* bl2     = (const float*)d_in[14];
    const float* beta3   = (const float*)d_in[15];
    const float* wl3     = (const float*)d_in[16];
    const float* bl3     = (const float*)d_in[17];
    const float* beta_li = (const float*)d_in[18];
    const float* w_out   = (const float*)d_in[19];
    const float* b_out   = (const float*)d_in[20];

    float* ws = (float*)d_ws;
    const size_t OFF_F3   = 0;
    const size_t OFF_F1   = 97028096;                 // 256*94*126*32 + 2048
    const size_t OFF_F2   = OFF_F1 + 26091520;        // + 256*98*130*8
    const size_t OFF_A    = OFF_F2;                   // f2 dead after conv3
    const size_t OFF_P    = OFF_F1;                   // f1 dead after conv2
    const size_t OFF_PART = OFF_F1 + 10560000;
    const size_t OFF_H    = OFF_PART + 262144;
    const size_t OFF_YS   = OFF_H + 8192;
    float* f1   = ws + OFF_F1;
    float* f2   = ws + OFF_F2;
    float* f3   = ws + OFF_F3;
    float* Abuf = ws + OFF_A;
    float* pbuf = ws + OFF_P;
    float* part = ws + OFF_PART;
    float* hbuf = ws + OFF_H;
    float* ybuf = ws + OFF_YS;

    conv1_kernel<<<101920, 256, 0, stream>>>(x, w1, b1, f1);
    conv2_kernel<<<24576, 256, 0, stream>>>(f1, w2, b2, f2);      // 196,608 waves
    conv3_kernel<<<24064, 256, 0, stream>>>(f2, w3, b3, f3);      // one (t,y) strip/block
    poolA_kernel<<<180480, 256, 0, stream>>>(f3, Abuf);
    poolB_kernel<<<41250, 256, 0, stream>>>(Abuf, pbuf);
    fc_kernel<<<128, 256, 0, stream>>>(pbuf, w_in, part);         // 1024 waves
    fc_reduce_kernel<<<32, 256, 0, stream>>>(part, b_in, hbuf);
    rnn_kernel<<<1, 1024, 0, stream>>>(hbuf, beta1, wl1, bl1, beta2, wl2, bl2,
                                       beta3, wl3, bl3, beta_li, ybuf);
    out_kernel<<<2, 256, 0, stream>>>(ybuf, w_out, b_out, (float*)d_out);
}